// GraphQNN_65481071404416
// MI455X (gfx1250) — compile-verified
//
#include <hip/hip_runtime.h>

// GraphQNN forward: per node (row of 16), three layers of
//   cur = cumprod(cos(cur + theta_l), axis=-1)
// Output = concat(x, act1, act2, act3) flat, f32.
//
// Bandwidth-bound: 64MB in + 256MB out => ~14us floor at 23.3 TB/s.
// Strategy: single pass, B128 non-temporal loads/stores, scalar(SGPR) params,
// v_cos_f32 fast cosine, gfx1250 global_prefetch_b8 to cover the grid-stride jump.

typedef float v4f __attribute__((ext_vector_type(4)));

#define N_WIRES  16
#define N_LAYERS 3

__global__ __launch_bounds__(256) void qnn_cumprod_cos_kernel(
    const float* __restrict__ x,
    const float* __restrict__ params,   // [N_LAYERS][N_WIRES]
    float* __restrict__ out,            // [4][n_nodes][N_WIRES]
    int n_nodes)
{
    // Uniform-address reads -> scalar loads into SGPRs (constant cache).
    float th[N_LAYERS][N_WIRES];
#pragma unroll
    for (int l = 0; l < N_LAYERS; ++l)
#pragma unroll
        for (int w = 0; w < N_WIRES; ++w)
            th[l][w] = params[l * N_WIRES + w];

    const unsigned stride = gridDim.x * blockDim.x;
    const size_t   seg    = (size_t)n_nodes * N_WIRES;   // one activation segment

    for (unsigned node = blockIdx.x * blockDim.x + threadIdx.x;
         node < (unsigned)n_nodes; node += stride) {

        // Speculative prefetch of this thread's next grid-stride row
        // (gfx1250 global_prefetch_b8; dropped silently if past the buffer).
        __builtin_prefetch(x + ((size_t)node + stride) * N_WIRES, 0, 1);

        const v4f* __restrict__ xr = (const v4f*)(x + (size_t)node * N_WIRES);
        v4f r[4];
#pragma unroll
        for (int q = 0; q < 4; ++q)
            r[q] = __builtin_nontemporal_load(xr + q);

        // activations[0] = x : echo from registers, write-once -> non-temporal.
        {
            v4f* __restrict__ o0 = (v4f*)(out + (size_t)node * N_WIRES);
#pragma unroll
            for (int q = 0; q < 4; ++q)
                __builtin_nontemporal_store(r[q], o0 + q);
        }

        float cur[N_WIRES];
#pragma unroll
        for (int q = 0; q < 4; ++q)
#pragma unroll
            for (int i = 0; i < 4; ++i)
                cur[q * 4 + i] = r[q][i];

#pragma unroll
        for (int l = 0; l < N_LAYERS; ++l) {
            // Nonlinear scan: serial cumprod in registers (15 v_mul_f32),
            // cosine via v_cos_f32 (args bounded, hardware range is ample).
            float prod = 1.0f;
#pragma unroll
            for (int w = 0; w < N_WIRES; ++w) {
                prod *= __cosf(cur[w] + th[l][w]);
                cur[w] = prod;
            }
            v4f* __restrict__ ol =
                (v4f*)(out + (size_t)(l + 1) * seg + (size_t)node * N_WIRES);
#pragma unroll
            for (int q = 0; q < 4; ++q) {
                v4f v = { cur[q * 4 + 0], cur[q * 4 + 1],
                          cur[q * 4 + 2], cur[q * 4 + 3] };
                __builtin_nontemporal_store(v, ol + q);
            }
        }
    }
}

extern "C" void kernel_launch(void* const* d_in, const int* in_sizes, int n_in,
                              void* d_out, int out_size, void* d_ws, size_t ws_size,
                              hipStream_t stream) {
    const float* x      = (const float*)d_in[0];   // (1e6, 16) f32
    const float* params = (const float*)d_in[1];   // (3, 16)  f32
    float*       out    = (float*)d_out;           // 4 * 1e6 * 16 f32

    const int n_nodes = in_sizes[0] / N_WIRES;

    // Grid-stride with ~2 rows per thread: 2048 blocks x 8 wave32 = 16384 waves,
    // enough to saturate every WGP while making the next-row prefetch useful.
    const int block = 256;
    const int grid  = 2048;
    qnn_cumprod_cos_kernel<<<grid, block, 0, stream>>>(x, params, out, n_nodes);
}